// GINEBlock_65034394796266
// MI455X (gfx1250) — compile-verified
//
#include <hip/hip_runtime.h>

#define N_NODES 40000
#define N_EDGES 640000
#define NODE_DIM 128
#define EDGE_DIM 64
#define HIDDEN_DIM 256
#define IN_DIM 192
#define LN_EPS 1e-5f

typedef float v2f __attribute__((ext_vector_type(2)));
typedef float v8f __attribute__((ext_vector_type(8)));

// padded LDS row strides (in floats) to break bank-conflict patterns
#define W1P 34    // sW1t[256][W1P]  (k-panel of 32, +2 pad)
#define W2P 260   // sW2t[128][W2P]  (k=256, +4 pad)
#define HP  258   // sH per-wave [16][HP] (k=256, +2 pad)
#define WUP 132   // sWut[128][WUP]

__device__ __forceinline__ v8f wmma_f32(v2f a, v2f b, v8f c) {
  // D = A(16x4 f32) * B(4x16 f32) + C(16x16 f32)
  return __builtin_amdgcn_wmma_f32_16x16x4_f32(false, a, false, b, (short)0, c,
                                               false, false);
}

// ---------------------------------------------------------------- kernel A
__global__ __launch_bounds__(256) void init_agg(const float4* __restrict__ x,
                                                float4* __restrict__ agg) {
  int i = blockIdx.x * 256 + threadIdx.x;  // 1,280,000 float4 total
  agg[i] = x[i];
}

// ---------------------------------------------------------------- kernel B
__global__ __launch_bounds__(256, 1)
void edge_mlp_scatter(const float* __restrict__ x, const float* __restrict__ ea,
                      const long long* __restrict__ eidx,
                      const float* __restrict__ W1, const float* __restrict__ b1,
                      const float* __restrict__ W2, const float* __restrict__ b2,
                      float* __restrict__ agg) {
  __shared__ float sW2t[128 * W2P];      // W2^T: [n][k]
  __shared__ float sW1t[256 * W1P];      // W1^T panel: [n][k_local]
  __shared__ float sH[8 * 16 * HP];      // per-wave h tiles [row][k]
  __shared__ float sB1[HIDDEN_DIM];
  __shared__ float sB2[NODE_DIM];
  __shared__ int sDst[8 * 16];

  const int t = threadIdx.x;
  const int wave = t >> 5, lane = t & 31;
  const int half = lane >> 4;      // 0 | 1
  const int sel = half << 1;       // 0 | 2  (K sub-selector for A/B frags)
  const int ln16 = lane & 15;

  // stage W2^T (once) + biases
  for (int i = 0; i < 128; ++i) {
    int idx = i * 256 + t;                       // coalesced read of W2[k][n]
    sW2t[(idx & 127) * W2P + (idx >> 7)] = W2[idx];
  }
  sB1[t] = b1[t];
  if (t < NODE_DIM) sB2[t] = b2[t];

  // edge indices for this wave's 16 edges
  const int ebase = (blockIdx.x * 8 + wave) * 16;
  const int e = ebase + ln16;
  const int srcn = (int)eidx[e];
  const int dstn = (int)eidx[N_EDGES + e];
  if (lane < 16) sDst[wave * 16 + lane] = dstn;

  __syncthreads();

  // h accumulators (16 N-tiles of 16), init with b1 broadcast per column
  v8f acc[16];
#pragma unroll
  for (int n = 0; n < 16; ++n) {
    float bv = sB1[n * 16 + ln16];
    acc[n] = (v8f){bv, bv, bv, bv, bv, bv, bv, bv};
  }

  const float* xrow = x + (long long)srcn * NODE_DIM;
  const float* earow = ea + (long long)e * EDGE_DIM;

  // GEMM1: h = m_in[16x192] @ W1[192x256], K-panels of 32
  for (int p = 0; p < 6; ++p) {
    __syncthreads();
    for (int i = 0; i < 32; ++i) {
      int idx = i * 256 + t;  // coalesced read of W1[k][n], transposed store
      int kk = idx >> 8, n = idx & 255;
      sW1t[n * W1P + kk] = W1[(p * 32 + kk) * 256 + n];
    }
    __syncthreads();

    v2f af[8];
#pragma unroll
    for (int kk = 0; kk < 8; ++kk) {
      int k0 = p * 32 + kk * 4 + sel;
      const float* ptr =
          (k0 < NODE_DIM) ? (xrow + k0) : (earow + (k0 - NODE_DIM));
      af[kk] = *(const v2f*)ptr;  // A frag: lane holds (k0, k0+1)
    }
#pragma unroll
    for (int n = 0; n < 16; ++n) {
#pragma unroll
      for (int kk = 0; kk < 8; ++kk) {
        v2f bf = *(const v2f*)&sW1t[(n * 16 + ln16) * W1P + kk * 4 + sel];
        acc[n] = wmma_f32(af[kk], bf, acc[n]);
      }
    }
  }

  // ReLU(h) -> per-wave LDS tile (same-wave dependency only; no barrier)
  const int hbase = wave * 16 * HP;
#pragma unroll
  for (int n = 0; n < 16; ++n) {
#pragma unroll
    for (int r = 0; r < 8; ++r) {
      float v = acc[n][r];
      v = v > 0.f ? v : 0.f;
      sH[hbase + (r + half * 8) * HP + n * 16 + ln16] = v;
    }
  }

  // GEMM2: m = h[16x256] @ W2[256x128]
  v8f acc2[8];
#pragma unroll
  for (int n = 0; n < 8; ++n) {
    float bv = sB2[n * 16 + ln16];
    acc2[n] = (v8f){bv, bv, bv, bv, bv, bv, bv, bv};
  }
  const int arow = hbase + ln16 * HP;
  for (int kb = 0; kb < 8; ++kb) {
    v2f a2[8];
#pragma unroll
    for (int kk = 0; kk < 8; ++kk)
      a2[kk] = *(const v2f*)&sH[arow + kb * 32 + kk * 4 + sel];
#pragma unroll
    for (int n = 0; n < 8; ++n) {
#pragma unroll
      for (int kk = 0; kk < 8; ++kk) {
        v2f bf = *(const v2f*)&sW2t[(n * 16 + ln16) * W2P + kb * 32 + kk * 4 + sel];
        acc2[n] = wmma_f32(a2[kk], bf, acc2[n]);
      }
    }
  }

  // ReLU(m) -> scatter-add into agg[dst]
  int drow[8];
#pragma unroll
  for (int r = 0; r < 8; ++r) drow[r] = sDst[wave * 16 + r + half * 8];
#pragma unroll
  for (int n = 0; n < 8; ++n) {
#pragma unroll
    for (int r = 0; r < 8; ++r) {
      float v = acc2[n][r];
      v = v > 0.f ? v : 0.f;
      atomicAdd(agg + (long long)drow[r] * NODE_DIM + n * 16 + ln16, v);
    }
  }
}

// ---------------------------------------------------------------- kernel C1
__global__ __launch_bounds__(256)
void ln_relu(const float* __restrict__ agg, const float* __restrict__ gamma,
             const float* __restrict__ beta, float* __restrict__ a2) {
  const int wave = threadIdx.x >> 5, lane = threadIdx.x & 31;
  const int node = blockIdx.x * 8 + wave;  // 5000 blocks exact
  float4 v = ((const float4*)(agg + (long long)node * NODE_DIM))[lane];
  float s = v.x + v.y + v.z + v.w;
  float sq = v.x * v.x + v.y * v.y + v.z * v.z + v.w * v.w;
#pragma unroll
  for (int o = 16; o > 0; o >>= 1) {
    s += __shfl_xor(s, o, 32);
    sq += __shfl_xor(sq, o, 32);
  }
  float mu = s * (1.f / NODE_DIM);
  float var = sq * (1.f / NODE_DIM) - mu * mu;
  float rs = rsqrtf(var + LN_EPS);
  float4 g = ((const float4*)gamma)[lane];
  float4 bt = ((const float4*)beta)[lane];
  float4 o4;
  o4.x = fmaxf((v.x - mu) * rs * g.x + bt.x, 0.f);
  o4.y = fmaxf((v.y - mu) * rs * g.y + bt.y, 0.f);
  o4.z = fmaxf((v.z - mu) * rs * g.z + bt.z, 0.f);
  o4.w = fmaxf((v.w - mu) * rs * g.w + bt.w, 0.f);
  ((float4*)(a2 + (long long)node * NODE_DIM))[lane] = o4;
}

// ---------------------------------------------------------------- kernel C2
__global__ __launch_bounds__(256, 1)
void node_gemm(const float* __restrict__ a2, const float* __restrict__ Wu,
               const float* __restrict__ bu, float* __restrict__ out) {
  __shared__ float sWut[128 * WUP];
  __shared__ float sBu[NODE_DIM];
  const int t = threadIdx.x;
  const int wave = t >> 5, lane = t & 31;
  const int half = lane >> 4, sel = half << 1, ln16 = lane & 15;

  for (int i = 0; i < 64; ++i) {
    int idx = i * 256 + t;
    sWut[(idx & 127) * WUP + (idx >> 7)] = Wu[idx];
  }
  if (t < NODE_DIM) sBu[t] = bu[t];
  __syncthreads();

  const int rows0 = (blockIdx.x * 8 + wave) * 16;
  int r0 = rows0 + ln16;
  int rc = r0 < N_NODES ? r0 : N_NODES - 1;  // clamp for tail loads
  const float* arow = a2 + (long long)rc * NODE_DIM;

  v8f acc[8];
#pragma unroll
  for (int n = 0; n < 8; ++n) {
    float bv = sBu[n * 16 + ln16];
    acc[n] = (v8f){bv, bv, bv, bv, bv, bv, bv, bv};
  }
  for (int kb = 0; kb < 4; ++kb) {
    v2f af[8];
#pragma unroll
    for (int kk = 0; kk < 8; ++kk)
      af[kk] = *(const v2f*)&arow[kb * 32 + kk * 4 + sel];
#pragma unroll
    for (int n = 0; n < 8; ++n) {
#pragma unroll
      for (int kk = 0; kk < 8; ++kk) {
        v2f bf = *(const v2f*)&sWut[(n * 16 + ln16) * WUP + kb * 32 + kk * 4 + sel];
        acc[n] = wmma_f32(af[kk], bf, acc[n]);
      }
    }
  }
#pragma unroll
  for (int n = 0; n < 8; ++n) {
#pragma unroll
    for (int r = 0; r < 8; ++r) {
      int gr = rows0 + r + half * 8;
      if (gr < N_NODES) out[(long long)gr * NODE_DIM + n * 16 + ln16] = acc[n][r];
    }
  }
}

// ---------------------------------------------------------------- launch
extern "C" void kernel_launch(void* const* d_in, const int* in_sizes, int n_in,
                              void* d_out, int out_size, void* d_ws,
                              size_t ws_size, hipStream_t stream) {
  (void)in_sizes; (void)n_in; (void)out_size; (void)ws_size;
  const float* x = (const float*)d_in[0];
  const float* ea = (const float*)d_in[1];
  const long long* eidx = (const long long*)d_in[2];
  const float* W1 = (const float*)d_in[3];
  const float* b1 = (const float*)d_in[4];
  const float* W2 = (const float*)d_in[5];
  const float* b2 = (const float*)d_in[6];
  const float* gamma = (const float*)d_in[7];
  const float* beta = (const float*)d_in[8];
  const float* Wu = (const float*)d_in[9];
  const float* bu = (const float*)d_in[10];
  float* outp = (float*)d_out;

  float* agg = (float*)d_ws;                                   // 40000*128 f32
  float* a2 = agg + (size_t)N_NODES * NODE_DIM;                // 40000*128 f32

  init_agg<<<5000, 256, 0, stream>>>((const float4*)x, (float4*)agg);
  edge_mlp_scatter<<<N_EDGES / 128, 256, 0, stream>>>(x, ea, eidx, W1, b1, W2,
                                                      b2, agg);
  ln_relu<<<N_NODES / 8, 256, 0, stream>>>(agg, gamma, beta, a2);
  node_gemm<<<(N_NODES + 127) / 128, 256, 0, stream>>>(a2, Wu, bu, outp);
}